// LocallyConnected2d_5970004542124
// MI455X (gfx1250) — compile-verified
//
#include <hip/hip_runtime.h>

typedef __attribute__((ext_vector_type(2))) float v2f;
typedef __attribute__((ext_vector_type(8))) float v8f;

#define B_   8
#define C_   32
#define H_   32
#define W_   32
#define O_   32
#define K2_  9
#define CCH  4                    // c-chunk (so K per chunk = 36, 9 WMMA steps)
#define KCH  (CCH * K2_)          // 36
#define NCHUNK (C_ / CCH)         // 8
#define NLOC 2                    // w locations per workgroup
#define WSTR 40                   // padded LDS stride for weight rows (floats)
#define ASTR 16                   // LDS stride for A rows (floats)
#define THREADS 128
#define LDSW_SZ (NLOC * KCH * WSTR)     // 2880 floats / buffer
#define LDSA_SZ (NLOC * KCH * ASTR)     // 1152 floats / buffer
#define NWELEM  (O_ * CCH * NLOC * K2_) // 2304 weight elems / chunk

#if defined(__AMDGCN__) && __has_builtin(__builtin_amdgcn_global_load_async_to_lds_b32)
#define HAS_ASYNC 1
#else
#define HAS_ASYNC 0
#endif

#if HAS_ASYNC
typedef __attribute__((address_space(1))) int g_i32;   // global ("__device__ int *")
typedef __attribute__((address_space(3))) int l_i32;   // LDS    ("__shared__ int *")
#endif

__device__ __forceinline__ void async_copy_b32(const float* g, float* l) {
#if HAS_ASYNC
    __builtin_amdgcn_global_load_async_to_lds_b32(
        (g_i32*)g, (l_i32*)l, /*offset=*/0, /*cpol=*/0);
#else
    *l = *g;   // synchronous fallback
#endif
}

__device__ __forceinline__ void wait_async_done() {
#if HAS_ASYNC && __has_builtin(__builtin_amdgcn_s_wait_asynccnt)
    __builtin_amdgcn_s_wait_asynccnt(0);
#endif
}

__global__ __launch_bounds__(THREADS)
void lc2d_wmma_async(const float* __restrict__ x,
                     const float* __restrict__ wgt,
                     float* __restrict__ out) {
    // Double-buffered LDS staging:
    //   lds_w[buf][loc][K][o]   (WMMA-B layout, padded stride 40 -> conflict-free halves)
    //   lds_a[buf][loc][K][b16] (WMMA-A layout, rows 8..15 and halo taps stay zero)
    __shared__ float lds_w[2][LDSW_SZ];   // 23040 B
    __shared__ float lds_a[2][LDSA_SZ];   //  9216 B

    const int tid  = threadIdx.x;
    const int wave = tid >> 5;
    const int lane = tid & 31;
    const int half = lane >> 4;          // 0: K pair {kk,kk+1}, 1: {kk+2,kk+3}
    const int n    = lane & 15;          // N index (and M index for A reads)

    const int h   = blockIdx.x >> 4;             // 0..31
    const int w0  = (blockIdx.x & 15) * NLOC;    // 0,2,...,30
    const int loc = wave >> 1;                   // which of the 2 w locations
    const int ot  = wave & 1;                    // o tile
    const int wl  = w0 + loc;

    // ---- zero A buffers once: pad rows / out-of-range taps are chunk-invariant ----
    for (int j = tid; j < 2 * LDSA_SZ; j += THREADS)
        ((float*)lds_a)[j] = 0.0f;
    __syncthreads();   // dscnt-waited barrier: zeros land before any async write

    // ---- async stage of one c-chunk into LDS buffer `buf` ----
    auto issue_chunk = [&](int c0, int buf) {
        // weights: runs of NLOC*9 = 18 contiguous floats per (o, c_local)
        for (int j = tid; j < NWELEM; j += THREADS) {
            const int r  = j / (NLOC * K2_);      // (o, c_local) run id
            const int e  = j - r * (NLOC * K2_);  // lw*9 + k within run
            const int o  = r >> 2;                // CCH == 4
            const int cl = r & 3;
            const int lw = e / K2_;
            const int k  = e - lw * K2_;
            const int kidx = cl * K2_ + k;
            const float* g =
                wgt + (((o * C_ + (c0 + cl)) * H_ + h) * W_ + w0) * K2_ + e;
            async_copy_b32(g, &lds_w[buf][(lw * KCH + kidx) * WSTR + o]);
        }
        // patches: only valid (b < 8, in-range) entries; the rest stay zero
        for (int j = tid; j < LDSA_SZ; j += THREADS) {
            const int b16  = j & 15;
            const int t    = j >> 4;              // loc*KCH + kidx
            const int kidx = t % KCH;
            const int lw   = t / KCH;
            const int cl   = kidx / K2_;
            const int k    = kidx - cl * K2_;
            const int dy   = k / 3 - 1;
            const int dx   = (k - (k / 3) * 3) - 1;
            const int hy   = h + dy;
            const int wx   = w0 + lw + dx;
            if (b16 < B_ && (unsigned)hy < (unsigned)H_ && (unsigned)wx < (unsigned)W_)
                async_copy_b32(x + ((b16 * C_ + (c0 + cl)) * H_ + hy) * W_ + wx,
                               &lds_a[buf][j]);
        }
    };

    issue_chunk(0, 0);   // prologue prefetch

    v8f acc = {};
    for (int ci = 0; ci < NCHUNK; ++ci) {
        const int buf = ci & 1;

        wait_async_done();   // this wave's prefetch of `buf` is complete
        __syncthreads();     // ... and every other wave's too

        // prefetch next chunk into the other buffer (its readers finished
        // before the barrier above), overlapping with the WMMA work below
        if (ci + 1 < NCHUNK)
            issue_chunk((ci + 1) * CCH, buf ^ 1);

        const float* arow = &lds_a[buf][(loc * KCH + 2 * half) * ASTR + n];
        const float* brow = &lds_w[buf][(loc * KCH + 2 * half) * WSTR + ot * 16 + n];

        #pragma unroll
        for (int kk = 0; kk < KCH; kk += 4) {
            v2f a, b;
            a.x = arow[kk * ASTR];
            a.y = arow[kk * ASTR + ASTR];
            b.x = brow[kk * WSTR];
            b.y = brow[kk * WSTR + WSTR];
            acc = __builtin_amdgcn_wmma_f32_16x16x4_f32(
                /*neg_a=*/false, a, /*neg_b=*/false, b,
                /*c_mod=*/(short)0, acc, /*reuse_a=*/false, /*reuse_b=*/false);
        }
    }

    // ---- writeback: lanes 0-15 hold M=0..7 (valid batches) in acc[0..7] ----
    if (lane < 16) {
        const int o = ot * 16 + n;
        float* op = out + (o * H_ + h) * W_ + wl;
        #pragma unroll
        for (int r = 0; r < B_; ++r)
            op[r * (O_ * H_ * W_)] = acc[r];
    }
}

extern "C" void kernel_launch(void* const* d_in, const int* in_sizes, int n_in,
                              void* d_out, int out_size, void* d_ws, size_t ws_size,
                              hipStream_t stream) {
    const float* x   = (const float*)d_in[0];   // (8, 32, 32, 32)
    const float* wgt = (const float*)d_in[1];   // (1, 32, 32, 32, 32, 9) -> flat
    float* out = (float*)d_out;                  // (8, 32, 32, 32)

    const dim3 grid(H_ * (W_ / NLOC));  // 512 workgroups: one per (h, w-pair)
    const dim3 block(THREADS);          // 4 waves: 2 locations x 2 o-tiles
    lc2d_wmma_async<<<grid, block, 0, stream>>>(x, wgt, out);
}